// BatchSuperQMulti_2808908612237
// MI455X (gfx1250) — compile-verified
//
#include <hip/hip_runtime.h>
#include <hip/hip_bf16.h>

typedef __attribute__((ext_vector_type(2))) float v2f;
typedef __attribute__((ext_vector_type(8))) float v8f;

#define SQ_EPS 1e-6f
#define INV_2PI 0.15915494309189535f
#define WAVES_PER_BLOCK 8
#define TILES_PER_WAVE  8
#define PTS_PER_BLOCK   (WAVES_PER_BLOCK * TILES_PER_WAVE * 16)

// ---- thin wrappers over the single-instruction AMDGPU transcendentals ----
__device__ __forceinline__ float fast_exp2(float x) { return __builtin_amdgcn_exp2f(x); }
__device__ __forceinline__ float fast_log2(float x) { return __builtin_amdgcn_logf(x); }
__device__ __forceinline__ float fast_sqrt(float x) { return __builtin_amdgcn_sqrtf(x); }
__device__ __forceinline__ float fast_rcp (float x) { return __builtin_amdgcn_rcpf(x); }
__device__ __forceinline__ float fast_rsq (float x) { return __builtin_amdgcn_rsqf(x); }
// v_sin_f32 / v_cos_f32 operate on turns: sin(2*pi*x)
__device__ __forceinline__ float fast_sin (float x) { return __builtin_amdgcn_sinf(x * INV_2PI); }
__device__ __forceinline__ float fast_cos (float x) { return __builtin_amdgcn_cosf(x * INV_2PI); }

// Minimax atan2: range-reduce to [0,1] with v_rcp, degree-9 odd polynomial
// (|err| < 1e-5 rad), quadrant fixups as selects. ~13 VALU ops vs ~35 for OCML.
__device__ __forceinline__ float fast_atan2(float y, float x) {
    float ax = fabsf(x), ay = fabsf(y);
    float mx = fmaxf(fmaxf(ax, ay), 1e-30f);   // guard atan2(0,0) -> 0
    float mn = fminf(ax, ay);
    float a  = mn * fast_rcp(mx);
    float s  = a * a;
    float p  = fmaf(s, 0.0208351f, -0.0851330f);
    p = fmaf(s, p, 0.1801410f);
    p = fmaf(s, p, -0.3302995f);
    p = fmaf(s, p, 0.9998660f);
    float r = a * p;
    r = (ay > ax) ? (1.57079632679f - r) : r;
    r = (x < 0.f) ? (3.14159265359f - r) : r;
    return __builtin_copysignf(r, y);
}

__device__ __forceinline__ float sgn_clamp(float v) {
    float s = (v > 0.f) ? 1.f : -1.f;
    return s * fmaxf(fabsf(v), SQ_EPS);
}

// safe_pow(x,p) = pow(max(|x|,EPS), p) via HW v_log_f32 / v_exp_f32
__device__ __forceinline__ float safe_pow(float v, float p) {
    return fast_exp2(p * fast_log2(fmaxf(fabsf(v), SQ_EPS)));
}

// In-place inverse bending on (u,v,w).
// Identity: sqrt(u^2+v^2) * cos(alpha - atan2(v,u)) == u*cos(alpha) + v*sin(alpha)
// removes one atan2, one sqrt and one cos per axis vs the literal reference form.
__device__ __forceinline__ void inv_bend(float& u, float& v, float& w,
                                         float inv_kb, float ca, float sa) {
    float Rr    = fmaf(u, ca, v * sa);
    float tt    = inv_kb - Rr;
    float gamma = fast_atan2(w, tt);
    float r     = inv_kb - fast_sqrt(fmaf(w, w, tt * tt));
    float dd    = Rr - r;
    u = fmaf(-dd, ca, u);
    v = fmaf(-dd, sa, v);
    w = inv_kb * gamma;
}

// W = R(q)^T (3x3), c = -R^T t  (so X = W*p + c)
__device__ __forceinline__ void quat_to_WT_c(const float* __restrict__ q4,
                                             const float* __restrict__ t3,
                                             float W[3][3], float c[3]) {
    float qw = q4[0], qx = q4[1], qy = q4[2], qz = q4[3];
    float inv = fast_rsq(qw * qw + qx * qx + qy * qy + qz * qz);
    qw *= inv; qx *= inv; qy *= inv; qz *= inv;
    float r00 = 1.f - 2.f * (qy * qy + qz * qz);
    float r01 = 2.f * (qx * qy - qw * qz);
    float r02 = 2.f * (qx * qz + qw * qy);
    float r10 = 2.f * (qx * qy + qw * qz);
    float r11 = 1.f - 2.f * (qx * qx + qz * qz);
    float r12 = 2.f * (qy * qz - qw * qx);
    float r20 = 2.f * (qx * qz - qw * qy);
    float r21 = 2.f * (qy * qz + qw * qx);
    float r22 = 1.f - 2.f * (qx * qx + qy * qy);
    // W[i][j] = R[j][i]
    W[0][0] = r00; W[0][1] = r10; W[0][2] = r20;
    W[1][0] = r01; W[1][1] = r11; W[1][2] = r21;
    W[2][0] = r02; W[2][1] = r12; W[2][2] = r22;
    c[0] = -(W[0][0] * t3[0] + W[0][1] * t3[1] + W[0][2] * t3[2]);
    c[1] = -(W[1][0] * t3[0] + W[1][1] * t3[1] + W[1][2] * t3[2]);
    c[2] = -(W[2][0] * t3[0] + W[2][1] * t3[1] + W[2][2] * t3[2]);
}

__global__ __launch_bounds__(WAVES_PER_BLOCK * 32)
void sq_sdf_wmma_kernel(const float* __restrict__ points,       // (B,3,M)
                        const float* __restrict__ scale,        // (B,N,3)
                        const float* __restrict__ exponents,    // (B,N,2)
                        const float* __restrict__ quat,         // (B,N,4)
                        const float* __restrict__ translation,  // (B,N,3)
                        const float* __restrict__ tapering,     // (B,N,2)
                        const float* __restrict__ bending,      // (B,N,6)
                        float* __restrict__ out,                // (B,N,M)
                        int B, int N, int M) {
    const int lane = threadIdx.x & 31;
    const int wave = threadIdx.x >> 5;
    const int half = lane >> 4;   // 0: primitive A rows 0-2, 1: primitive B rows 8-10
    const int li   = lane & 15;

    const int b      = blockIdx.z;
    const int pair   = blockIdx.y;       // primitive pair index
    const int nA     = pair * 2;
    const int n_lane = nA + half;        // this lane's primitive

    // ---- per-pair rotation fragments (uniform across the wave) ----
    float WA[3][3], WB[3][3], cA[3], cB[3];
    quat_to_WT_c(quat + (size_t)(b * N + nA) * 4,
                 translation + (size_t)(b * N + nA) * 3, WA, cA);
    quat_to_WT_c(quat + (size_t)(b * N + nA + 1) * 4,
                 translation + (size_t)(b * N + nA + 1) * 3, WB, cB);

    // A (16x4 f32): lane<16 holds row li {K0,K1}, lane>=16 holds row li {K2,K3}.
    // Rows 0-2 = R_A^T, rows 8-10 = R_B^T, rest zero (K=3 column is zero pad).
    v2f afrag; afrag.x = 0.f; afrag.y = 0.f;
    {
        int row = li;
        const float(*Wsel)[3] = nullptr;
        int i = 0;
        if (row < 3)                    { Wsel = WA; i = row; }
        else if (row >= 8 && row < 11)  { Wsel = WB; i = row - 8; }
        if (Wsel) {
            afrag.x = half ? Wsel[i][2] : Wsel[i][0];   // K2 : K0
            afrag.y = half ? 0.f        : Wsel[i][1];   // K3 : K1
        }
    }

    // C accumulator: row r gets -R^T t broadcast across all 16 columns.
    v8f cfrag = {};
    cfrag[0] = half ? cB[0] : cA[0];
    cfrag[1] = half ? cB[1] : cA[1];
    cfrag[2] = half ? cB[2] : cA[2];

    // ---- per-lane primitive parameters (hoisted out of the point loop) ----
    const size_t pidx = (size_t)(b * N + n_lane);
    const float* sc = scale + pidx * 3;
    const float inv_sx = 1.f / sc[0], inv_sy = 1.f / sc[1], inv_sz = 1.f / sc[2];
    const float* ex = exponents + pidx * 2;
    const float e1 = ex[0], e2 = ex[1];
    const float ie1 = 1.f / e1, ie2 = 1.f / e2, e21 = e2 * ie1, ne1h = -0.5f * e1;
    const float* tp = tapering + pidx * 2;
    const float kxs = tp[0] * inv_sz, kys = tp[1] * inv_sz;
    const float* bd = bending + pidx * 6;
    const float ikb0 = 1.f / (bd[0] + 1e-6f), al0 = bd[1];
    const float ikb1 = 1.f / (bd[2] + 1e-6f), al1 = bd[3];
    const float ikb2 = 1.f / (bd[4] + 1e-6f), al2 = bd[5];
    const float ca0 = fast_cos(al0), sa0 = fast_sin(al0);
    const float ca1 = fast_cos(al1), sa1 = fast_sin(al1);
    const float ca2 = fast_cos(al2), sa2 = fast_sin(al2);

    const float* pb  = points + (size_t)b * 3 * M;
    const int koff   = half ? 2 * M : 0;   // B VGPR0 carries x-row (K0) | z-row (K2)
    float* outrow    = out + pidx * M;

    const int m_base = blockIdx.x * PTS_PER_BLOCK + wave * (TILES_PER_WAVE * 16);

    for (int t = 0; t < TILES_PER_WAVE; ++t) {
        const int m0 = m_base + t * 16;
        if (m0 >= M) break;                       // wave-uniform exit
        const int mi = m0 + li;
        const int mc = (mi < M) ? mi : (M - 1);   // clamp, keep EXEC all-ones

        // B (4x16 f32): VGPR0 = {K0 | K2}, VGPR1 = {K1 | K3(=0)}
        v2f bfrag;
        bfrag.x = pb[koff + mc];                  // x (lanes 0-15) / z (lanes 16-31)
        {
            float yv = pb[M + mc];
            bfrag.y = half ? 0.f : yv;            // y / zero-pad
        }

        if (t + 1 < TILES_PER_WAVE && m0 + 32 <= M)
            __builtin_prefetch(&pb[koff + m0 + 16], 0, 0);  // global_prefetch_b8

        // X = R^T p - R^T t for 2 primitives x 16 points; result lands per-lane.
        v8f d = __builtin_amdgcn_wmma_f32_16x16x4_f32(
            false, afrag, false, bfrag, (short)0, cfrag, false, false);

        float x = sgn_clamp(d[0]);
        float y = sgn_clamp(d[1]);
        float z = sgn_clamp(d[2]);

        // tapering
        x *= fast_rcp(sgn_clamp(kxs * z + 1.f));
        y *= fast_rcp(sgn_clamp(kys * z + 1.f));

        // inverse bending: z-axis, x-axis, y-axis (reference permutation order)
        inv_bend(x, y, z, ikb0, ca0, sa0);
        { float u = y, v = z, w = x; inv_bend(u, v, w, ikb1, ca1, sa1); x = w; y = u; z = v; }
        { float u = z, v = x, w = y; inv_bend(u, v, w, ikb2, ca2, sa2); x = v; y = w; z = u; }

        const float r0 = fast_sqrt(x * x + y * y + z * z);

        float tx = fmaxf(fabsf(x * inv_sx), SQ_EPS);
        float ty = fmaxf(fabsf(y * inv_sy), SQ_EPS);
        float tz = fmaxf(fabsf(z * inv_sz), SQ_EPS);
        const float term1 = safe_pow(tx * tx, ie2);
        const float term2 = safe_pow(ty * ty, ie2);
        const float term3 = safe_pow(tz * tz, ie1);
        const float f   = safe_pow(safe_pow(term1 + term2, e21) + term3, ne1h);
        const float sdf = r0 * (1.f - f);

        if (mi < M) outrow[mi] = sdf;             // 2x 64B coalesced segments
    }
}

extern "C" void kernel_launch(void* const* d_in, const int* in_sizes, int n_in,
                              void* d_out, int out_size, void* d_ws, size_t ws_size,
                              hipStream_t stream) {
    (void)n_in; (void)out_size; (void)d_ws; (void)ws_size;
    const int B = 4;
    const int N = 32;
    const int M = in_sizes[0] / (B * 3);   // points is (B,3,M)

    const float* points      = (const float*)d_in[0];
    const float* scale       = (const float*)d_in[1];
    const float* exponents   = (const float*)d_in[2];
    const float* quat        = (const float*)d_in[3];
    const float* translation = (const float*)d_in[4];
    const float* tapering    = (const float*)d_in[5];
    const float* bending     = (const float*)d_in[6];
    float* out               = (float*)d_out;

    dim3 block(WAVES_PER_BLOCK * 32);
    dim3 grid((M + PTS_PER_BLOCK - 1) / PTS_PER_BLOCK, N / 2, B);
    hipLaunchKernelGGL(sq_sdf_wmma_kernel, grid, block, 0, stream,
                       points, scale, exponents, quat, translation, tapering,
                       bending, out, B, N, M);
}